// ONNXSAGEConv_70583492543063
// MI455X (gfx1250) — compile-verified
//
#include <hip/hip_runtime.h>
#include <hip/hip_bf16.h>

typedef float v2f __attribute__((ext_vector_type(2)));
typedef float v8f __attribute__((ext_vector_type(8)));

#define FEAT 64  // D_IN == D_OUT == 64

// ---------------------------------------------------------------------------
// Kernel 1: zero the accumulator (d_out) with wide stores.
// ---------------------------------------------------------------------------
__global__ __launch_bounds__(256) void sage_zero_kernel(float4* __restrict__ p, long long n4) {
    long long i = (long long)blockIdx.x * blockDim.x + threadIdx.x;
    if (i < n4) p[i] = make_float4(0.f, 0.f, 0.f, 0.f);
}

// ---------------------------------------------------------------------------
// Kernel 2: edge scatter.  16 threads cooperate per edge: each loads a float4
// slice of x[src] (coalesced 256B row read, L2-resident) and does 4
// global_atomic_add_f32 into acc[dst] (L2 atomic units, no-return).
// ---------------------------------------------------------------------------
__global__ __launch_bounds__(256) void sage_scatter_kernel(const float* __restrict__ x,
                                                           const int* __restrict__ src,
                                                           const int* __restrict__ dst,
                                                           float* __restrict__ acc,
                                                           long long n_edges) {
    long long t = (long long)blockIdx.x * blockDim.x + threadIdx.x;
    long long e = t >> 4;
    if (e >= n_edges) return;
    const int f = ((int)t & 15) * 4;
    const int s = src[e];
    const int d = dst[e];
    const float4 v = *(const float4*)(x + (size_t)s * FEAT + f);
    float* p = acc + (size_t)d * FEAT + f;
    atomicAdd(p + 0, v.x);
    atomicAdd(p + 1, v.y);
    atomicAdd(p + 2, v.z);
    atomicAdd(p + 3, v.w);
}

// ---------------------------------------------------------------------------
// Kernel 3: fused mean + GEMM + bias, IN PLACE on acc (= d_out).
//   out[m][n] = (acc[m][:] / deg[m]) . W[n][:] + bias[n]
// One wave computes a 16(M) x 64(N) tile with V_WMMA_F32_16X16X4_F32,
// K = 64 in 16 steps.  W is staged in LDS once per block.
//
// Operand layouts (CDNA5 ISA 7.12.2, wave32, g = lane/16, idx = lane%16):
//   A 16x4 f32 : lane holds {x[idx][k0+2g], x[idx][k0+2g+1]}        (v2f)
//   B 4x16 f32 : lane holds {W[idx+16t][k0+2g], W[idx+16t][k0+2g+1]} (v2f)
//   C/D 16x16  : VGPR v -> row v+8g, col idx                        (v8f)
// In-place safety: a wave reads only rows [m_base, m_base+16); all loads are
// issued (and waited on for the WMMAs) before the epilogue stores; waves own
// disjoint row ranges.
// ---------------------------------------------------------------------------
__global__ __launch_bounds__(256) void sage_gemm_kernel(float* __restrict__ acc,
                                                        const float* __restrict__ deg,
                                                        const float* __restrict__ W,
                                                        const float* __restrict__ bias,
                                                        int n_nodes) {
    __shared__ float Wlds[FEAT * FEAT];  // 16 KB (320 KB LDS per WGP: plenty)
    for (int i = threadIdx.x; i < FEAT * FEAT; i += blockDim.x) Wlds[i] = W[i];
    __syncthreads();

    const int wave = threadIdx.x >> 5;  // 8 waves per block
    const int lane = threadIdx.x & 31;
    const int g    = lane >> 4;   // half-wave group
    const int idx  = lane & 15;

    const int m_base = (blockIdx.x * 8 + wave) * 16;
    if (m_base >= n_nodes) return;  // wave-uniform exit; EXEC stays all-1s

    // Row this lane feeds into the A operand (clamped for safety; N%16==0 here).
    int m = m_base + idx;
    if (m >= n_nodes) m = n_nodes - 1;
    const float rinv = 1.0f / deg[m];                  // fused mean
    const float* arow = acc + (size_t)m * FEAT;

    // Seed accumulators with the bias (each D column n gets bias[n]).
    v8f c0, c1, c2, c3;
    {
        const float b0 = bias[idx +  0];
        const float b1 = bias[idx + 16];
        const float b2 = bias[idx + 32];
        const float b3 = bias[idx + 48];
#pragma unroll
        for (int v = 0; v < 8; ++v) { c0[v] = b0; c1[v] = b1; c2[v] = b2; c3[v] = b3; }
    }

#pragma unroll
    for (int k0 = 0; k0 < FEAT; k0 += 4) {
        const int k = k0 + 2 * g;
        v2f a = *(const v2f*)(arow + k);
        a *= rinv;  // mean fused into A (row-scaling commutes with the GEMM)
        const v2f b0 = *(const v2f*)(&Wlds[(idx +  0) * FEAT + k]);
        const v2f b1 = *(const v2f*)(&Wlds[(idx + 16) * FEAT + k]);
        const v2f b2 = *(const v2f*)(&Wlds[(idx + 32) * FEAT + k]);
        const v2f b3 = *(const v2f*)(&Wlds[(idx + 48) * FEAT + k]);
        // 8 args: (neg_a, A, neg_b, B, c_mod, C, reuse_a, reuse_b)
        c0 = __builtin_amdgcn_wmma_f32_16x16x4_f32(false, a, false, b0, (short)0, c0, false, false);
        c1 = __builtin_amdgcn_wmma_f32_16x16x4_f32(false, a, false, b1, (short)0, c1, false, false);
        c2 = __builtin_amdgcn_wmma_f32_16x16x4_f32(false, a, false, b2, (short)0, c2, false, false);
        c3 = __builtin_amdgcn_wmma_f32_16x16x4_f32(false, a, false, b3, (short)0, c3, false, false);
    }

    // Epilogue: D VGPR v holds row (m_base + v + 8g), column (idx + 16*tile).
    const int row0 = m_base + 8 * g;
#pragma unroll
    for (int v = 0; v < 8; ++v) {
        const int r = row0 + v;
        if (r < n_nodes) {
            float* orow = acc + (size_t)r * FEAT + idx;
            orow[ 0] = c0[v];
            orow[16] = c1[v];
            orow[32] = c2[v];
            orow[48] = c3[v];
        }
    }
}

// ---------------------------------------------------------------------------
// Launch: inputs in setup_inputs() order:
//   d_in[0]=x [N,64] f32, d_in[1]=edge_index [2,E] int, d_in[2]=node_degrees
//   [N] f32, d_in[3]=W [64,64] f32, d_in[4]=b [64] f32.  d_out: [N,64] f32.
// ---------------------------------------------------------------------------
extern "C" void kernel_launch(void* const* d_in, const int* in_sizes, int n_in,
                              void* d_out, int out_size, void* d_ws, size_t ws_size,
                              hipStream_t stream) {
    (void)n_in; (void)d_ws; (void)ws_size;

    const float* x    = (const float*)d_in[0];
    const int*   ei   = (const int*)d_in[1];
    const float* deg  = (const float*)d_in[2];
    const float* W    = (const float*)d_in[3];
    const float* bias = (const float*)d_in[4];
    float*       out  = (float*)d_out;

    const int       n_nodes = in_sizes[2];
    const long long n_edges = (long long)in_sizes[1] / 2;
    const int* src = ei;                 // edge_index[0, :]
    const int* dst = ei + n_edges;       // edge_index[1, :]

    // 1) zero the accumulator (d_out is poisoned / stale between replays)
    {
        const long long n4 = (long long)out_size / 4;
        const int blocks = (int)((n4 + 255) / 256);
        sage_zero_kernel<<<blocks, 256, 0, stream>>>((float4*)out, n4);
    }
    // 2) scatter-add x[src] into out[dst]
    {
        const long long threads = n_edges * 16;
        const int blocks = (int)((threads + 255) / 256);
        sage_scatter_kernel<<<blocks, 256, 0, stream>>>(x, src, dst, out, n_edges);
    }
    // 3) fused mean + WMMA GEMM + bias, in place
    {
        const int blocks = (n_nodes + 127) / 128;  // 8 waves x 16 rows per block
        sage_gemm_kernel<<<blocks, 256, 0, stream>>>(out, deg, W, bias, n_nodes);
    }
}